// GCNEncoder_39822936769196
// MI455X (gfx1250) — compile-verified
//
#include <hip/hip_runtime.h>
#include <hip/hip_bf16.h>

// ---------------------------------------------------------------------------
// GraphSAGE (2 layers, mean aggregation) for MI455X / gfx1250.
//   Layer: out = relu?( (mean_{j->i} x_j) @ W_l + b + x_i @ W_r )
// Dominant cost: edge gather + scatter-add (L2-resident f32 atomics).
// GEMMs use full-precision V_WMMA_F32_16X16X4_F32 (wave32 matrix pipe).
// ---------------------------------------------------------------------------

typedef __attribute__((ext_vector_type(2))) float v2f;
typedef __attribute__((ext_vector_type(8))) float v8f;

// --- degree count: one thread per edge ------------------------------------
__global__ void sage_count_kernel(const int* __restrict__ dst,
                                  int* __restrict__ cnt, int E) {
    int e = blockIdx.x * blockDim.x + threadIdx.x;
    if (e < E) atomicAdd(&cnt[dst[e]], 1);
}

// --- scatter aggregation: one wave (32 lanes) per edge, float4 per lane ----
__global__ void sage_aggregate_kernel(const float* __restrict__ feat,
                                      const int* __restrict__ src,
                                      const int* __restrict__ dst,
                                      float* __restrict__ agg,
                                      int E, int d) {
    int wavesPerBlock = blockDim.x >> 5;
    int e = blockIdx.x * wavesPerBlock + (threadIdx.x >> 5);
    if (e >= E) return;
    int lane = threadIdx.x & 31;
    long long s = src[e];
    long long t = dst[e];
    const float4* xs = (const float4*)(feat + s * (long long)d);
    float* ap = agg + t * (long long)d;
    int d4 = d >> 2;                       // float4s per row (32 when d==128)
    for (int c = lane; c < d4; c += 32) {
        float4 v = xs[c];
        float* p = ap + c * 4;
        atomicAdd(p + 0, v.x);
        atomicAdd(p + 1, v.y);
        atomicAdd(p + 2, v.z);
        atomicAdd(p + 3, v.w);
    }
}

// --- scale rows by 1/max(deg,1) --------------------------------------------
__global__ void sage_scale_kernel(float* __restrict__ agg,
                                  const int* __restrict__ cnt,
                                  int total4, int d4) {
    int i = blockIdx.x * blockDim.x + threadIdx.x;
    if (i >= total4) return;
    int row = i / d4;
    int c = cnt[row];
    float inv = 1.0f / (float)(c > 0 ? c : 1);
    float4* p = (float4*)agg;
    float4 v = p[i];
    v.x *= inv; v.y *= inv; v.z *= inv; v.w *= inv;
    p[i] = v;
}

// --- fused dual-GEMM with f32 WMMA -----------------------------------------
// out[m, n] = relu?( sum_k Aagg[m,k] Wl[k,n] + sum_k Ax[m,k] Wr[k,n] + b[n] )
// One wave computes one 16x16 output tile.
// A frag (16x4 f32, 2 VGPRs): lane l -> M = l%16; K-pair base = (l<16)?0:2.
// B frag (4x16 f32, 2 VGPRs): lane l -> N = l%16; K-pair base = (l<16)?0:2.
// C/D (8 VGPRs): lane l -> N = l%16; VGPR v -> M = v + (l<16 ? 0 : 8).
__global__ __launch_bounds__(32)
void sage_gemm_wmma(const float* __restrict__ Aagg,
                    const float* __restrict__ Ax,
                    const float* __restrict__ Wl,
                    const float* __restrict__ Wr,
                    const float* __restrict__ bias,
                    float* __restrict__ out,
                    int M, int K, int Nout, int relu) {
    int tileM = blockIdx.x;
    int tileN = blockIdx.y;
    int lane = threadIdx.x;

    int mf = tileM * 16 + (lane & 15);            // row this lane loads for A
    int n  = tileN * 16 + (lane & 15);            // col this lane owns in B/C/D
    int kh = (lane >> 4) << 1;                    // 0 or 2: K sub-pair base
    int mload = mf < M ? mf : (M - 1);            // clamp (output is guarded)

    const float* arow_agg = Aagg + (long long)mload * K;
    const float* arow_x   = Ax   + (long long)mload * K;

    v8f acc = {};
    for (int k = 0; k < K; k += 4) {
        v2f a, b;
        a.x = arow_agg[k + kh];
        a.y = arow_agg[k + kh + 1];
        b.x = Wl[(long long)(k + kh)     * Nout + n];
        b.y = Wl[(long long)(k + kh + 1) * Nout + n];
        acc = __builtin_amdgcn_wmma_f32_16x16x4_f32(
            false, a, false, b, (short)0, acc, false, false);
    }
    for (int k = 0; k < K; k += 4) {
        v2f a, b;
        a.x = arow_x[k + kh];
        a.y = arow_x[k + kh + 1];
        b.x = Wr[(long long)(k + kh)     * Nout + n];
        b.y = Wr[(long long)(k + kh + 1) * Nout + n];
        acc = __builtin_amdgcn_wmma_f32_16x16x4_f32(
            false, a, false, b, (short)0, acc, false, false);
    }

    float bv = bias[n];
    int mbase = tileM * 16 + ((lane >> 4) << 3);  // +0 or +8
    #pragma unroll
    for (int v = 0; v < 8; ++v) {
        int mrow = mbase + v;
        if (mrow < M) {
            float val = acc[v] + bv;
            if (relu) val = fmaxf(val, 0.0f);
            out[(long long)mrow * Nout + n] = val;
        }
    }
}

// ---------------------------------------------------------------------------
extern "C" void kernel_launch(void* const* d_in, const int* in_sizes, int n_in,
                              void* d_out, int out_size, void* d_ws, size_t ws_size,
                              hipStream_t stream) {
    const float* x   = (const float*)d_in[0];
    const int*   ei  = (const int*)d_in[1];   // edge_index [2, E]
    const float* W1l = (const float*)d_in[2];
    const float* b1  = (const float*)d_in[3];
    const float* W1r = (const float*)d_in[4];
    const float* W2l = (const float*)d_in[5];
    const float* b2  = (const float*)d_in[6];
    const float* W2r = (const float*)d_in[7];
    float* out = (float*)d_out;

    const int HID  = in_sizes[3];             // 128
    const int OUTC = in_sizes[6];             // 64
    const int INC  = in_sizes[2] / HID;       // 128
    const int N    = in_sizes[0] / INC;       // 50000
    const int E    = in_sizes[1] / 2;         // 800000

    const int* src = ei;
    const int* dst = ei + E;

    // workspace carve-out
    char* ws = (char*)d_ws;
    size_t off = 0;
    auto carve = [&](size_t bytes) {
        char* p = ws + off;
        off = (off + bytes + 255) & ~(size_t)255;
        return p;
    };
    int*   cnt  = (int*)  carve((size_t)N * sizeof(int));
    float* agg1 = (float*)carve((size_t)N * INC * sizeof(float));
    float* h    = (float*)carve((size_t)N * HID * sizeof(float));
    float* agg2 = (float*)carve((size_t)N * HID * sizeof(float));

    hipMemsetAsync(cnt,  0, (size_t)N * sizeof(int),          stream);
    hipMemsetAsync(agg1, 0, (size_t)N * INC * sizeof(float),  stream);
    hipMemsetAsync(agg2, 0, (size_t)N * HID * sizeof(float),  stream);

    // shared degree counts
    sage_count_kernel<<<(E + 255) / 256, 256, 0, stream>>>(dst, cnt, E);

    // ---- layer 1 ----
    sage_aggregate_kernel<<<(E + 7) / 8, 256, 0, stream>>>(x, src, dst, agg1, E, INC);
    {
        int d4 = INC / 4, total4 = N * d4;
        sage_scale_kernel<<<(total4 + 255) / 256, 256, 0, stream>>>(agg1, cnt, total4, d4);
    }
    {
        dim3 grid((N + 15) / 16, HID / 16);
        sage_gemm_wmma<<<grid, 32, 0, stream>>>(agg1, x, W1l, W1r, b1, h,
                                                N, INC, HID, /*relu=*/1);
    }

    // ---- layer 2 ----
    sage_aggregate_kernel<<<(E + 7) / 8, 256, 0, stream>>>(h, src, dst, agg2, E, HID);
    {
        int d4 = HID / 4, total4 = N * d4;
        sage_scale_kernel<<<(total4 + 255) / 256, 256, 0, stream>>>(agg2, cnt, total4, d4);
    }
    {
        dim3 grid((N + 15) / 16, OUTC / 16);
        sage_gemm_wmma<<<grid, 32, 0, stream>>>(agg2, h, W2l, W2r, b2, out,
                                                N, HID, OUTC, /*relu=*/0);
    }
}